// CorrelationLayer_15221364097237
// MI455X (gfx1250) — compile-verified
//
#include <hip/hip_runtime.h>

// CorrelationLayer for MI455X (gfx1250, wave32).
//  out[b,d,h,w] = (dot - 256*mu1*mu2) / (s1*s2),
//  dot = sum_c x1[b,c,h,w] * x2pad[b,c,h+2d,w],  x2pad = zero-pad(20) rows/cols,
//  accessed w column of x2pad is (w) in padded coords == original w-20.
// Kernel 1: per-position mean / unbiased std for x1 and the accessed x2pad region -> d_ws.
// Kernel 2: band-GEMM via V_WMMA_F32_16X16X4_F32 (16h x 64r tiles per (b,w)), LDS-staged.
// Workspace requirement: 86016 floats = 344064 bytes.

typedef float v2f __attribute__((ext_vector_type(2)));
typedef float v8f __attribute__((ext_vector_type(8)));

#define B_    4
#define C_    256
#define H_    64
#define W_    64
#define PAD_  20
#define ND_   21          // displacements (d = 0..20, row offset 2d-20 .. handled in padded coords)
#define RP_   104         // padded row count
#define CINV  (1.0f/256.0f)
#define VINV  (1.0f/255.0f)   // unbiased (ddof=1)

// ws layout in floats
#define MU1_OFF 0
#define S1_OFF  16384
#define MU2_OFF 32768
#define S2_OFF  59392

// ---------------- Kernel 1: stats ----------------
__global__ __launch_bounds__(256)
void corr_stats(const float* __restrict__ x1, const float* __restrict__ x2,
                float* __restrict__ ws) {
  const int P1 = B_ * H_ * W_;      // 16384
  const int P2 = B_ * RP_ * W_;     // 26624
  int gid = blockIdx.x * 256 + threadIdx.x;
  if (gid < P1) {
    int b  = gid >> 12;             // H_*W_ = 4096
    int hw = gid & 4095;
    const float* p = x1 + (size_t)b * (C_ * H_ * W_) + hw;
    float s = 0.f, q = 0.f;
#pragma unroll 4
    for (int c = 0; c < C_; ++c) { float v = p[(size_t)c * 4096]; s += v; q += v * v; }
    float var = fmaxf((q - s * s * CINV) * VINV, 0.f);
    ws[MU1_OFF + gid] = s * CINV;
    ws[S1_OFF  + gid] = sqrtf(var);
  } else if (gid < P1 + P2) {
    int g   = gid - P1;
    int b   = g / (RP_ * W_);
    int rem = g - b * (RP_ * W_);
    int r   = rem >> 6;             // padded row 0..103
    int wp  = rem & 63;             // padded col 0..63
    int oh = r - PAD_, ow = wp - PAD_;
    float s = 0.f, q = 0.f;
    if (oh >= 0 && oh < H_ && ow >= 0 && ow < W_) {
      const float* p = x2 + (size_t)b * (C_ * H_ * W_) + oh * W_ + ow;
#pragma unroll 4
      for (int c = 0; c < C_; ++c) { float v = p[(size_t)c * 4096]; s += v; q += v * v; }
    }
    float var = fmaxf((q - s * s * CINV) * VINV, 0.f);
    ws[MU2_OFF + g] = s * CINV;
    ws[S2_OFF  + g] = sqrtf(var);
  }
}

// ---------------- Kernel 2: WMMA band-GEMM ----------------
#define KC 16             // c-chunk depth
#define WT 4              // w values per block (1 per wave)
#define X1_ROW   5        // 16 rows * stride 5 (conflict-free: gcd(5,64)=1)
#define X1_PLANE 80       // 16*5
#define X2_ROW   5
#define X2_PLANE 321      // 64*5 + 1 (plane stride != 0 mod 64 -> half-waves conflict-free)
#define X1_OFF 0
#define X2_OFF (KC * X1_PLANE)                       // 1280
#define LDS_FLOATS (KC * X1_PLANE + KC * X2_PLANE)   // 6416 floats = 25664 B

__global__ __launch_bounds__(128)
void corr_main(const float* __restrict__ x1, const float* __restrict__ x2,
               const float* __restrict__ ws, float* __restrict__ out) {
  extern __shared__ float lds[];
  const int tid = threadIdx.x;
  const int bid = blockIdx.x;                  // grid = 4b * 4ht * 16wq = 256
  const int wq = bid & 15;
  const int ht = (bid >> 4) & 3;
  const int b  = bid >> 6;
  const int h0 = ht * 16;
  const int w0 = wq * WT;

  const int lane = tid & 31;
  const int wav  = tid >> 5;                   // 0..3 => local w
  const int mrow = lane & 15;                  // M (or N) index within fragment
  const int kh   = lane >> 4;                  // half-wave: K pair select

  v8f acc[4] = {};                             // 4 r-tiles of 16, covering r-h0 in [0,64)
  const size_t xbase = (size_t)b * (C_ * H_ * W_);

  for (int c0 = 0; c0 < C_; c0 += KC) {
    // stage x1: kc(16) x m(16) x wl(4), coalesced along w
    for (int idx = tid; idx < KC * 16 * WT; idx += 128) {
      int wl = idx & (WT - 1);
      int m  = (idx >> 2) & 15;
      int kc = idx >> 6;
      float v = x1[xbase + (size_t)(c0 + kc) * 4096 + (h0 + m) * 64 + (w0 + wl)];
      lds[X1_OFF + kc * X1_PLANE + m * X1_ROW + wl] = v;
    }
    // stage x2pad: kc(16) x r(64) x wl(4); rows >=104 / pad region -> 0
    for (int idx = tid; idx < KC * 64 * WT; idx += 128) {
      int wl = idx & (WT - 1);
      int rl = (idx >> 2) & 63;
      int kc = idx >> 8;
      int oh = h0 + rl - PAD_;
      int ow = w0 + wl - PAD_;
      float v = 0.f;
      if (oh >= 0 && oh < H_ && ow >= 0 && ow < W_)
        v = x2[xbase + (size_t)(c0 + kc) * 4096 + oh * 64 + ow];
      lds[X2_OFF + kc * X2_PLANE + rl * X2_ROW + wl] = v;
    }
    if (c0 + KC < C_) {                        // gfx1250 global_prefetch_b8 for next chunk
      __builtin_prefetch(&x1[xbase + (size_t)(c0 + KC) * 4096 + h0 * 64 + w0], 0, 1);
      __builtin_prefetch(&x2[xbase + (size_t)(c0 + KC) * 4096], 0, 1);
    }
    __syncthreads();

#pragma unroll
    for (int k4 = 0; k4 < KC / 4; ++k4) {
      const int kb = k4 * 4 + 2 * kh;          // ISA A-layout: lanes 0-15 K0/K1, 16-31 K2/K3
      v2f a;
      a.x = lds[X1_OFF + (kb    ) * X1_PLANE + mrow * X1_ROW + wav];
      a.y = lds[X1_OFF + (kb + 1) * X1_PLANE + mrow * X1_ROW + wav];
#pragma unroll
      for (int t = 0; t < 4; ++t) {
        v2f bb;                                // B: K rows striped across lanes, N = lane&15
        bb.x = lds[X2_OFF + (kb    ) * X2_PLANE + (t * 16 + mrow) * X2_ROW + wav];
        bb.y = lds[X2_OFF + (kb + 1) * X2_PLANE + (t * 16 + mrow) * X2_ROW + wav];
        acc[t] = __builtin_amdgcn_wmma_f32_16x16x4_f32(
            false, a, false, bb, (short)0, acc[t], false, false);
      }
    }
    __syncthreads();
  }

  // mean/std correction, stage D tiles into LDS (reused) for w-contiguous stores
  const int   w   = w0 + wav;
  const float* mu1 = ws + MU1_OFF;
  const float* s1  = ws + S1_OFF;
  const float* mu2 = ws + MU2_OFF;
  const float* s2  = ws + S2_OFF;

#pragma unroll
  for (int t = 0; t < 4; ++t) {
#pragma unroll
    for (int i = 0; i < 8; ++i) {
      int m     = i + 8 * kh;                  // C/D layout: vgpr i, halves hold M=i / M=i+8
      int rl    = t * 16 + mrow;
      int delta = rl - m;                      // r - h
      if (delta >= 0 && delta <= 40 && (delta & 1) == 0) {
        int d = delta >> 1;
        int h = h0 + m;
        int r = h0 + rl;                       // <= 103 always for valid d
        float m1 = mu1[(b * H_  + h) * W_ + w];
        float g1 = s1 [(b * H_  + h) * W_ + w];
        float m2 = mu2[(b * RP_ + r) * W_ + w];
        float g2 = s2 [(b * RP_ + r) * W_ + w];
        float num = acc[t][i] - 256.0f * m1 * m2;
        lds[(d * 16 + m) * WT + wav] = num / (g1 * g2);   // 0/0 -> NaN matches reference
      }
    }
  }
  __syncthreads();
  for (int idx = tid; idx < ND_ * 16 * WT; idx += 128) {
    int wl = idx & (WT - 1);
    int p  = idx >> 2;                         // d*16 + hl
    int d  = p >> 4;
    int hl = p & 15;
    out[(((size_t)b * ND_ + d) * H_ + (h0 + hl)) * W_ + (w0 + wl)] = lds[p * WT + wl];
  }
}

extern "C" void kernel_launch(void* const* d_in, const int* in_sizes, int n_in,
                              void* d_out, int out_size, void* d_ws, size_t ws_size,
                              hipStream_t stream) {
  const float* x1 = (const float*)d_in[0];
  const float* x2 = (const float*)d_in[1];
  float* ws  = (float*)d_ws;      // needs >= 344064 bytes
  float* out = (float*)d_out;     // 4*21*64*64 = 344064 floats

  corr_stats<<<(16384 + 26624 + 255) / 256, 256, 0, stream>>>(x1, x2, ws);
  corr_main<<<256, 128, LDS_FLOATS * sizeof(float), stream>>>(x1, x2, ws, out);
}